// GraphValueHead_34488587387631
// MI455X (gfx1250) — compile-verified
//
#include <hip/hip_runtime.h>
#include <stdint.h>

typedef __attribute__((ext_vector_type(2))) float v2f;
typedef __attribute__((ext_vector_type(8))) float v8f;

#define N_NODES 50000
#define N_EDGES 800000
#define HID 64
#define EDGE_DIM 16
#define HEADS 4
#define HEAD_DIM 16
#define NEG_SLOPE 0.2f

// ---- float <-> order-preserving unsigned key (zero is the minimum key) ----
__device__ __forceinline__ unsigned fkey(float f) {
    unsigned b = __float_as_uint(f);
    return (b & 0x80000000u) ? ~b : (b | 0x80000000u);
}
__device__ __forceinline__ float fdecode(unsigned k) {
    return (k & 0x80000000u) ? __uint_as_float(k ^ 0x80000000u)
                             : __uint_as_float(~k);
}

// ---------------------------------------------------------------------------
// C[nrows x 64] = A[nrows x 64] * B[64 x 64] (+ bias), f32 WMMA 16x16x4.
// One wave computes a 16x64 tile: 4 v8f accumulators, K-loop of 16 WMMA steps
// per N-tile. Wave-uniform guard keeps EXEC all-ones for WMMA.
// ---------------------------------------------------------------------------
__global__ __launch_bounds__(256)
void gemm64_wmma_kernel(const float* __restrict__ A, const float* __restrict__ B,
                        const float* __restrict__ bias, float* __restrict__ C,
                        int nrows) {
    const int wave = threadIdx.x >> 5;
    const int lane = threadIdx.x & 31;
    const int tile = blockIdx.x * (blockDim.x >> 5) + wave;
    const int ntiles = nrows >> 4;           // nrows % 16 == 0 for all our GEMMs
    if (tile >= ntiles) return;              // uniform across the wave

    const int row0  = tile << 4;
    const int lrow  = lane & 15;
    const int khalf = (lane >> 4) << 1;      // lanes 0-15 -> K{0,1}; 16-31 -> K{2,3}

    v8f acc0 = {}; v8f acc1 = {}; v8f acc2 = {}; v8f acc3 = {};

    const float* Arow = A + (size_t)(row0 + lrow) * HID;
#pragma unroll
    for (int kb = 0; kb < HID; kb += 4) {
        v2f a;
        a.x = Arow[kb + khalf + 0];
        a.y = Arow[kb + khalf + 1];
        const float* B0 = B + (size_t)(kb + khalf) * HID + lrow;
        v2f b0; b0.x = B0[0];        b0.y = B0[HID];
        v2f b1; b1.x = B0[16];       b1.y = B0[HID + 16];
        v2f b2; b2.x = B0[32];       b2.y = B0[HID + 32];
        v2f b3; b3.x = B0[48];       b3.y = B0[HID + 48];
        acc0 = __builtin_amdgcn_wmma_f32_16x16x4_f32(false, a, false, b0, (short)0, acc0, false, false);
        acc1 = __builtin_amdgcn_wmma_f32_16x16x4_f32(false, a, false, b1, (short)0, acc1, false, false);
        acc2 = __builtin_amdgcn_wmma_f32_16x16x4_f32(false, a, false, b2, (short)0, acc2, false, false);
        acc3 = __builtin_amdgcn_wmma_f32_16x16x4_f32(false, a, false, b3, (short)0, acc3, false, false);
    }

    // C/D layout: component i -> row = row0 + i + 8*(lane/16), col = ntile*16 + lane%16
    const int rbase = row0 + ((lane >> 4) << 3);
    v8f accs[4] = {acc0, acc1, acc2, acc3};
#pragma unroll
    for (int nt = 0; nt < 4; ++nt) {
        const int col = (nt << 4) + lrow;
        const float bv = bias ? bias[col] : 0.0f;
#pragma unroll
        for (int i = 0; i < 8; ++i) {
            C[(size_t)(rbase + i) * HID + col] = accs[nt][i] + bv;
        }
    }
}

// ---- per-node attention coefficients: asrc/adst [N, HEADS] --------------
__global__ void node_att_kernel(const float* __restrict__ h,
                                const float* __restrict__ a_src,
                                const float* __restrict__ a_dst,
                                float* __restrict__ asrc, float* __restrict__ adst) {
    int t = blockIdx.x * blockDim.x + threadIdx.x;
    if (t >= N_NODES * HEADS) return;
    int n = t >> 2, hd = t & 3;
    const float* hp = h + (size_t)n * HID + hd * HEAD_DIM;
    const float* as = a_src + hd * HEAD_DIM;
    const float* ad = a_dst + hd * HEAD_DIM;
    float s = 0.f, d = 0.f;
#pragma unroll
    for (int i = 0; i < HEAD_DIM; ++i) { float v = hp[i]; s += v * as[i]; d += v * ad[i]; }
    asrc[t] = s; adst[t] = d;
}

// ---- ve[k][head] = sum_d w_edge[k, head*16+d] * a_edge[head, d] ----------
__global__ void edge_vec_kernel(const float* __restrict__ w_edge,
                                const float* __restrict__ a_edge,
                                float* __restrict__ ve) {
    int t = threadIdx.x;           // 64 threads: (k, head)
    int k = t >> 2, hd = t & 3;
    float s = 0.f;
#pragma unroll
    for (int d = 0; d < HEAD_DIM; ++d)
        s += w_edge[k * HID + hd * HEAD_DIM + d] * a_edge[hd * HEAD_DIM + d];
    ve[t] = s;
}

__global__ void zero_u32_kernel(unsigned* __restrict__ p, int n) {
    int t = blockIdx.x * blockDim.x + threadIdx.x;
    if (t < n) p[t] = 0u;
}

// ---- edge pass 1: logits + segment max (via ordered-uint atomicMax) ------
__global__ void edge_logits_kernel(const float* __restrict__ edge_attr,
                                   const int* __restrict__ ei,
                                   const float* __restrict__ asrc,
                                   const float* __restrict__ adst,
                                   const float* __restrict__ ve,
                                   float* __restrict__ logits,
                                   unsigned* __restrict__ mkey) {
    int e = blockIdx.x * blockDim.x + threadIdx.x;
    if (e >= N_EDGES) return;
    int s = ei[e], d = ei[N_EDGES + e];
    const float4* eap = (const float4*)(edge_attr + (size_t)e * EDGE_DIM);
    float4 q0 = eap[0], q1 = eap[1], q2 = eap[2], q3 = eap[3];
    float ea[16] = {q0.x,q0.y,q0.z,q0.w, q1.x,q1.y,q1.z,q1.w,
                    q2.x,q2.y,q2.z,q2.w, q3.x,q3.y,q3.z,q3.w};
#pragma unroll
    for (int hd = 0; hd < HEADS; ++hd) {
        float ae = 0.f;
#pragma unroll
        for (int k = 0; k < EDGE_DIM; ++k) ae += ea[k] * ve[k * HEADS + hd];
        float l = asrc[s * HEADS + hd] + adst[d * HEADS + hd] + ae;
        l = (l > 0.f) ? l : NEG_SLOPE * l;           // leaky relu
        logits[(size_t)e * HEADS + hd] = l;
        atomicMax(&mkey[d * HEADS + hd], fkey(l));
    }
}

// ---- edge pass 2: ex = exp(logit - m[dst]); den[dst] += ex ---------------
__global__ void edge_exp_kernel(const int* __restrict__ ei,
                                const unsigned* __restrict__ mkey,
                                float* __restrict__ exbuf,   // in: logits, out: ex
                                float* __restrict__ den) {
    int e = blockIdx.x * blockDim.x + threadIdx.x;
    if (e >= N_EDGES) return;
    int d = ei[N_EDGES + e];
#pragma unroll
    for (int hd = 0; hd < HEADS; ++hd) {
        float m = fdecode(mkey[d * HEADS + hd]);
        float ex = __expf(exbuf[(size_t)e * HEADS + hd] - m);
        exbuf[(size_t)e * HEADS + hd] = ex;
        atomicAdd(&den[d * HEADS + hd], ex);
    }
}

// ---- edge pass 3: agg[dst] += h[src] * attn (one thread per edge-head) ---
__global__ void edge_agg_kernel(const int* __restrict__ ei,
                                const float* __restrict__ exbuf,
                                const float* __restrict__ den,
                                const float* __restrict__ hbuf,
                                float* __restrict__ agg) {
    int t = blockIdx.x * blockDim.x + threadIdx.x;
    if (t >= N_EDGES * HEADS) return;
    int e = t >> 2, hd = t & 3;
    int s = ei[e], d = ei[N_EDGES + e];
    float attn = exbuf[t] / (den[d * HEADS + hd] + 1e-16f);
    const float* hp = hbuf + (size_t)s * HID + hd * HEAD_DIM;
    float* ap = agg + (size_t)d * HID + hd * HEAD_DIM;
#pragma unroll
    for (int i = 0; i < HEAD_DIM; ++i) atomicAdd(&ap[i], hp[i] * attn);
}

// ---- residual + relu ------------------------------------------------------
__global__ void residual_relu_kernel(float* __restrict__ x, const float* __restrict__ agg) {
    int t = blockIdx.x * blockDim.x + threadIdx.x;
    if (t >= N_NODES * HID) return;
    float v = x[t] + agg[t];
    x[t] = (v > 0.f) ? v : 0.f;
}

// ---- per-column sum & max over nodes -------------------------------------
__global__ void col_reduce_kernel(const float* __restrict__ x,
                                  float* __restrict__ gsum, unsigned* __restrict__ gmaxk) {
    int col = threadIdx.x;            // 64 threads per block
    int r0 = blockIdx.x * 256;
    int r1 = r0 + 256; if (r1 > N_NODES) r1 = N_NODES;
    float s = 0.f, m = -3.402823466e38f;
    for (int r = r0; r < r1; ++r) { float v = x[(size_t)r * HID + col]; s += v; m = fmaxf(m, v); }
    atomicAdd(&gsum[col], s);
    atomicMax(&gmaxk[col], fkey(m));
}

// ---- final MLP: scalar output --------------------------------------------
__global__ void final_mlp_kernel(const float* __restrict__ gsum,
                                 const unsigned* __restrict__ gmaxk,
                                 const float* __restrict__ w1, const float* __restrict__ b1,
                                 const float* __restrict__ w2, const float* __restrict__ b2,
                                 float* __restrict__ out) {
    __shared__ float emb[2 * HID];
    __shared__ float hm[HID];
    int t = threadIdx.x;              // 128 threads
    emb[t] = (t < HID) ? gsum[t] * (1.0f / (float)N_NODES) : fdecode(gmaxk[t - HID]);
    __syncthreads();
    if (t < HID) {
        float s = b1[t];
#pragma unroll 4
        for (int i = 0; i < 2 * HID; ++i) s += emb[i] * w1[i * HID + t];
        hm[t] = (s > 0.f) ? s : 0.f;
    }
    __syncthreads();
    if (t == 0) {
        float s = b2[0];
        for (int j = 0; j < HID; ++j) s += hm[j] * w2[j];
        out[0] = s;
    }
}

extern "C" void kernel_launch(void* const* d_in, const int* in_sizes, int n_in,
                              void* d_out, int out_size, void* d_ws, size_t ws_size,
                              hipStream_t stream) {
    const float* node_x    = (const float*)d_in[0];
    const float* edge_attr = (const float*)d_in[1];
    const int*   edge_idx  = (const int*)  d_in[2];
    const float* in_w      = (const float*)d_in[3];
    const float* in_b      = (const float*)d_in[4];
    const float* w_node    = (const float*)d_in[5];   // [3,64,64]
    const float* w_edge    = (const float*)d_in[6];   // [3,16,64]
    const float* a_src     = (const float*)d_in[7];   // [3,4,16]
    const float* a_dst     = (const float*)d_in[8];
    const float* a_edge    = (const float*)d_in[9];
    const float* mlp_w1    = (const float*)d_in[10];  // [128,64]
    const float* mlp_b1    = (const float*)d_in[11];
    const float* mlp_w2    = (const float*)d_in[12];  // [64,1]
    const float* mlp_b2    = (const float*)d_in[13];
    float* out = (float*)d_out;

    // ---- workspace layout (agg|mkey|den contiguous so one zero pass clears all)
    float*    x     = (float*)d_ws;                       // N*64
    float*    h     = x    + (size_t)N_NODES * HID;       // N*64
    float*    agg   = h    + (size_t)N_NODES * HID;       // N*64
    unsigned* mkey  = (unsigned*)(agg + (size_t)N_NODES * HID); // N*4
    float*    den   = (float*)(mkey + (size_t)N_NODES * HEADS); // N*4
    float*    exbuf = den  + (size_t)N_NODES * HEADS;     // E*4 (logits, then ex)
    float*    asrc  = exbuf + (size_t)N_EDGES * HEADS;    // N*4
    float*    adst  = asrc + (size_t)N_NODES * HEADS;     // N*4
    float*    ve    = adst + (size_t)N_NODES * HEADS;     // 64
    float*    gsum  = ve + 64;                            // 64
    unsigned* gmaxk = (unsigned*)(gsum + HID);            // 64

    const int ntiles = N_NODES / 16;                      // 3125
    const int gemm_blocks = (ntiles + 7) / 8;             // 8 waves/block

    // 1) input projection: x = node_x @ in_w + in_b
    gemm64_wmma_kernel<<<gemm_blocks, 256, 0, stream>>>(node_x, in_w, in_b, x, N_NODES);

    for (int l = 0; l < 3; ++l) {
        const float* wn = w_node + (size_t)l * HID * HID;
        const float* we = w_edge + (size_t)l * EDGE_DIM * HID;
        const float* as = a_src  + (size_t)l * HEADS * HEAD_DIM;
        const float* ad = a_dst  + (size_t)l * HEADS * HEAD_DIM;
        const float* ae = a_edge + (size_t)l * HEADS * HEAD_DIM;

        // h = x @ w_node[l]
        gemm64_wmma_kernel<<<gemm_blocks, 256, 0, stream>>>(x, wn, nullptr, h, N_NODES);
        // per-node attention coefficients
        node_att_kernel<<<(N_NODES * HEADS + 255) / 256, 256, 0, stream>>>(h, as, ad, asrc, adst);
        // folded edge-attention vector
        edge_vec_kernel<<<1, 64, 0, stream>>>(we, ae, ve);
        // clear agg | mkey | den (contiguous N*72 dwords)
        {
            int nz = N_NODES * (HID + 2 * HEADS);
            zero_u32_kernel<<<(nz + 255) / 256, 256, 0, stream>>>((unsigned*)agg, nz);
        }
        // edge passes
        edge_logits_kernel<<<(N_EDGES + 255) / 256, 256, 0, stream>>>(
            edge_attr, edge_idx, asrc, adst, ve, exbuf, mkey);
        edge_exp_kernel<<<(N_EDGES + 255) / 256, 256, 0, stream>>>(edge_idx, mkey, exbuf, den);
        edge_agg_kernel<<<(N_EDGES * HEADS + 255) / 256, 256, 0, stream>>>(
            edge_idx, exbuf, den, h, agg);
        // x = relu(x + agg)
        residual_relu_kernel<<<(N_NODES * HID + 255) / 256, 256, 0, stream>>>(x, agg);
    }

    // graph embedding: column mean & max, then MLP
    zero_u32_kernel<<<1, 128, 0, stream>>>((unsigned*)gsum, 2 * HID);
    col_reduce_kernel<<<(N_NODES + 255) / 256, 64, 0, stream>>>(x, gsum, gmaxk);
    final_mlp_kernel<<<1, 128, 0, stream>>>(gsum, gmaxk, mlp_w1, mlp_b1, mlp_w2, mlp_b2, out);
}